// UpSample1d_1726576854265
// MI455X (gfx1250) — compile-verified
//
#include <hip/hip_runtime.h>
#include <cstdint>

typedef __attribute__((ext_vector_type(2))) float v2f;
typedef __attribute__((ext_vector_type(8))) float v8f;

#define T_IN     8192
#define T_OUT    16384
#define ROWS     4096          // B*C
#define SPG      256           // input sample pairs per workgroup -> 512 outputs
#define LOAD_W   272           // floats staged per row (16B-aligned window)
#define LDS_W    276           // padded LDS row stride in floats (bank-conflict-free)

#define HAS_WMMA_F32 __has_builtin(__builtin_amdgcn_wmma_f32_16x16x4_f32)

// G[i][t]: contribution of xp[s_tile + i] to out[2*s_tile + t] (includes 2x gain)
__device__ __forceinline__ float gmat(const float* __restrict__ filt, int i, int t) {
    int r = t >> 1;
    int d = i - r;
    if (t & 1) {
        if (d >= 3 && d <= 8) return 2.0f * filt[16 - 2 * d];
    } else {
        if (d >= 2 && d <= 7) return 2.0f * filt[15 - 2 * d];
    }
    return 0.0f;
}

__global__ __launch_bounds__(256) void upsample2x_wmma_kernel(
    const float* __restrict__ x, const float* __restrict__ filt,
    float* __restrict__ out)
{
    __shared__ float tile[16 * LDS_W];   // 17,664 B

    const int tid  = threadIdx.x;
    const int lane = tid & 31;
    const int wave = tid >> 5;
    const int hi   = lane >> 4;          // lane half (selects K pair / D row half)
    const int tcol = lane & 15;          // N column / A row

    const int tg      = blockIdx.x;      // time group (32)
    const int rg      = blockIdx.y;      // row group (256)
    const int s0      = tg * SPG;        // first input pair index of this group
    const int c0      = s0 - 8;          // first staged global column (16B aligned)
    const int rowbase = rg * 16;

    // ---------- stage x[rowbase..+16)[c0..c0+LOAD_W) into LDS ----------
    // LDS col u+3 holds xp[s0+u] == x[s0+u-5] (interior)
    const bool interior = (c0 >= 0) && (c0 + LOAD_W <= T_IN);
    if (interior) {
        // async global->LDS, 68 b128 chunks per row, 1088 total
        for (int i = tid; i < 16 * 68; i += 256) {
            int r = i / 68;
            int c = i - r * 68;
            const float* gp = x + ((size_t)(rowbase + r) * T_IN + (size_t)(c0 + 4 * c));
            uint32_t loff = (uint32_t)(uintptr_t)(&tile[r * LDS_W + 4 * c]);
            uint64_t gaddr = (uint64_t)(uintptr_t)gp;
            asm volatile("global_load_async_to_lds_b128 %0, %1, off"
                         :: "v"(loff), "v"(gaddr) : "memory");
        }
        asm volatile("s_wait_asynccnt 0x0" ::: "memory");
    } else {
        // boundary time groups: clamped (edge-replicate) scalar staging
        for (int i = tid; i < 16 * LOAD_W; i += 256) {
            int r = i / LOAD_W;
            int c = i - r * LOAD_W;
            int g = c0 + c;
            g = g < 0 ? 0 : (g > T_IN - 1 ? T_IN - 1 : g);
            tile[r * LDS_W + c] = x[(size_t)(rowbase + r) * T_IN + (size_t)g];
        }
    }
    __syncthreads();

    // ---------- per-lane constant filter-band fragments ----------
#if HAS_WMMA_F32
    // B 4x16 f32 layout: VGPR j, lane L -> K = j + 2*(L>=16), N = L%16
    v2f bfrag[4];
    #pragma unroll
    for (int s = 0; s < 4; ++s) {
        bfrag[s].x = gmat(filt, 4 * s + 2 * hi + 0, tcol);
        bfrag[s].y = gmat(filt, 4 * s + 2 * hi + 1, tcol);
    }
#else
    float gcol[16];
    #pragma unroll
    for (int i = 0; i < 16; ++i) gcol[i] = gmat(filt, i, tcol);
#endif

    const size_t obase = (size_t)rowbase * T_OUT + (size_t)2 * s0;

    // ---------- 4 tiles per wave: D[16ch x 16out] = A[16x16] * G[16x16] ----------
    #pragma unroll
    for (int q = 0; q < 4; ++q) {
        const int tt      = wave + 8 * q;     // tile 0..31
        const int colbase = 3 + 8 * tt;       // LDS col of xp[s_tile]
        v8f acc = {};

#if HAS_WMMA_F32
        // A 16x4 f32 layout: lane L -> M = L%16; VGPR j -> K = j + 2*(L>=16)
        const int arow = tcol * LDS_W;
        #pragma unroll
        for (int s = 0; s < 4; ++s) {
            int cidx = arow + colbase + 4 * s + 2 * hi;
            v2f a;
            a.x = tile[cidx];
            a.y = tile[cidx + 1];
            acc = __builtin_amdgcn_wmma_f32_16x16x4_f32(
                false, a, false, bfrag[s], (short)0, acc, false, false);
        }
#else
        #pragma unroll
        for (int v = 0; v < 8; ++v) {
            const int m = v + 8 * hi;
            float sum = 0.0f;
            #pragma unroll
            for (int i = 0; i < 16; ++i)
                sum += tile[m * LDS_W + colbase + i] * gcol[i];
            acc[v] = sum;
        }
#endif

        // D layout: VGPR v, lanes 0-15 -> row v, lanes 16-31 -> row v+8; col = lane%16
        const int t0 = 16 * tt;
        #pragma unroll
        for (int v = 0; v < 8; ++v) {
            out[obase + (size_t)(v + 8 * hi) * T_OUT + (size_t)(t0 + tcol)] = acc[v];
        }
    }
}

extern "C" void kernel_launch(void* const* d_in, const int* in_sizes, int n_in,
                              void* d_out, int out_size, void* d_ws, size_t ws_size,
                              hipStream_t stream) {
    const float* x    = (const float*)d_in[0];   // [8, 512, 8192] fp32
    const float* filt = (const float*)d_in[1];   // [1, 1, 12]    fp32
    float*       out  = (float*)d_out;           // [8, 512, 16384] fp32

    dim3 grid(T_IN / SPG, ROWS / 16);            // (32, 256)
    upsample2x_wmma_kernel<<<grid, dim3(256), 0, stream>>>(x, filt, out);
}